// capsule_fusion_62156766708253
// MI455X (gfx1250) — compile-verified
//
#include <hip/hip_runtime.h>
#include <math.h>

// ---------------- problem constants ----------------
#define LN    32    // L
#define NTOK  1024  // N
#define DDIM  768   // D
#define KOUT  16    // N_OUT
#define IDIM  64    // IN_DIM
#define ODIM  64    // OUT_DIM
#define EPSF  1e-8f
#define HALF_PI 1.57079632679489662f

#define XPAD  772   // 768 + 4 : bank-conflict-free A reads (stride%64 == 4)
#define MUPAD 68    // 64 + 4
#define VPAD  68    // 64 + 4 : keeps per-lane rows 16B-aligned for async b128 -> LDS
#define MPAD  65    // mu/var rows: conflict-free k-indexed reads

typedef float v8f __attribute__((ext_vector_type(8)));
typedef float v2f __attribute__((ext_vector_type(2)));

// D(16x16,f32) += A(16x4,f32) x B(4x16,f32) ; full f32 precision matrix op.
#define WMMA4(a, b, c) \
    __builtin_amdgcn_wmma_f32_16x16x4_f32(false, (a), false, (b), (short)0, (c), false, false)

__device__ __forceinline__ float gelu_exact(float v) {
    return 0.5f * v * (1.0f + erff(v * 0.70710678118654752f));
}

// =====================================================================
// Kernel 1: fused  mu_in = gelu(x_t @ Wcap + Bcap)  ->  V = mu_in @ Wvote + Bvote
//           plus   f_a[l,n] = sigmoid(x_t @ Wscore + Bscore) * mask
// grid.x = L * (N/16);  block = 128 (4 waves, wave32)
// V stored as V[n][l][k][o]  (so kernel 2 reads one contiguous 128 KB chunk)
// =====================================================================
__global__ __launch_bounds__(128)
void k_stage1(const float* __restrict__ x,      // (N,L,D)
              const float* __restrict__ mask,   // (N,L)
              const float* __restrict__ Wscore, // (L,D,1)
              const float* __restrict__ Bscore, // (L,1,1)
              const float* __restrict__ Wcap,   // (L,D,IDIM)
              const float* __restrict__ Bcap,   // (L,1,IDIM)
              const float* __restrict__ Wvote,  // (L,KOUT,IDIM,ODIM)
              const float* __restrict__ Bvote,  // (L,KOUT,1,ODIM)
              float* __restrict__ fa_ws,        // (L,N)
              float* __restrict__ V_ws)         // (N,L,KOUT,ODIM)
{
    const int l  = blockIdx.x >> 6;
    const int n0 = (blockIdx.x & 63) * 16;

    __shared__ float xs[16 * XPAD];     // 16 rows of x (n-tile), padded
    __shared__ float mus[16 * MUPAD];   // gelu'd capsule inputs, padded

    const int tid  = threadIdx.x;
    const int lane = tid & 31;
    const int w    = tid >> 5;
    const int m    = lane & 15;   // row inside 16-wide dimension
    const int khi  = lane >> 4;   // which K half (ISA A-layout: lanes16-31 hold K=2,3)

    // ---- stage x tile (16 x 768) into LDS, coalesced float4 loads ----
    for (int idx = tid; idx < 16 * 192; idx += 128) {
        const int r  = idx / 192;
        const int c4 = idx % 192;
        const float4 v = *reinterpret_cast<const float4*>(
            x + ((size_t)(n0 + r) * LN + l) * DDIM + c4 * 4);
        float* d = xs + r * XPAD + c4 * 4;
        d[0] = v.x; d[1] = v.y; d[2] = v.z; d[3] = v.w;
    }
    __syncthreads();

    // ---- part A: mu tile (16n x 64i); wave w owns i-block o0 = 16*w ----
    const int o0 = w * 16;
    v8f acc = {};
    for (int k0 = 0; k0 < DDIM; k0 += 4) {
        const int ci = k0 + 2 * khi;
        v2f a, b;
        a.x = xs[m * XPAD + ci];
        a.y = xs[m * XPAD + ci + 1];
        const float* wc = Wcap + ((size_t)l * DDIM + ci) * IDIM + o0 + m;
        b.x = wc[0];
        b.y = wc[IDIM];
        acc = WMMA4(a, b, acc);
    }
    // epilogue: bias + exact gelu -> LDS (D layout: vgpr r = row r / r+8)
    for (int r = 0; r < 8; ++r) {
        const int mm = r + 8 * khi;
        const int oo = o0 + m;
        mus[mm * MUPAD + oo] = gelu_exact(acc[r] + Bcap[l * IDIM + oo]);
    }

    // ---- score path: f_a[l, n0+m] (threads 0..15, after WMMA is done) ----
    if (tid < 16) {
        const float* wsr = Wscore + (size_t)l * DDIM;
        float s = 0.f;
        for (int d = 0; d < DDIM; ++d) s += xs[tid * XPAD + d] * wsr[d];
        s += Bscore[l];
        const float sig = 1.f / (1.f + expf(-s));
        fa_ws[l * NTOK + n0 + tid] = sig * mask[(size_t)(n0 + tid) * LN + l];
    }
    __syncthreads();

    // ---- part B: votes V[l,k, n-tile, 0..63]; wave w handles k = w,w+4,w+8,w+12 ----
    for (int k = w; k < KOUT; k += 4) {
        v8f aj0 = {}, aj1 = {}, aj2 = {}, aj3 = {};
        const float* wv_base = Wvote + (((size_t)l * KOUT + k) * IDIM) * ODIM;
        for (int k0 = 0; k0 < IDIM; k0 += 4) {
            const int ci = k0 + 2 * khi;
            v2f a;
            a.x = mus[m * MUPAD + ci];
            a.y = mus[m * MUPAD + ci + 1];
            const float* wv = wv_base + (size_t)ci * ODIM;
            v2f b;
            b.x = wv[0 * 16 + m];  b.y = wv[ODIM + 0 * 16 + m];  aj0 = WMMA4(a, b, aj0);
            b.x = wv[1 * 16 + m];  b.y = wv[ODIM + 1 * 16 + m];  aj1 = WMMA4(a, b, aj1);
            b.x = wv[2 * 16 + m];  b.y = wv[ODIM + 2 * 16 + m];  aj2 = WMMA4(a, b, aj2);
            b.x = wv[3 * 16 + m];  b.y = wv[ODIM + 3 * 16 + m];  aj3 = WMMA4(a, b, aj3);
        }
        const float* bv = Bvote + ((size_t)l * KOUT + k) * ODIM;
        for (int r = 0; r < 8; ++r) {
            const int mm = r + 8 * khi;
            float* vout = V_ws + (((size_t)(n0 + mm) * LN + l) * KOUT + k) * ODIM;
            vout[0 * 16 + m] = aj0[r] + bv[0 * 16 + m];
            vout[1 * 16 + m] = aj1[r] + bv[1 * 16 + m];
            vout[2 * 16 + m] = aj2[r] + bv[2 * 16 + m];
            vout[3 * 16 + m] = aj3[r] + bv[3 * 16 + m];
        }
    }
}

// =====================================================================
// Kernel 2: EM routing, one workgroup per n.  V slice (128 KB) staged into
// LDS once via CDNA5 async global->LDS B128 DMA, then all routing
// iterations run out of LDS.  V fits the 192 MB L2 globally.
// grid.x = N ; block = 512 (16 waves) ; dynamic LDS ~152 KB -> 2 WGs/WGP.
// =====================================================================
__global__ __launch_bounds__(512)
void k_route(const float* __restrict__ fa_ws,   // (L,N)
             const float* __restrict__ V_ws,    // (N,L,KOUT,ODIM)
             const float* __restrict__ beta_use,// (L)
             const float* __restrict__ beta_ign,// (KOUT)
             const int*   __restrict__ iters_p,
             float* __restrict__ out)           // [N*KOUT] a_fin ++ [N*KOUT*ODIM] mu_fin
{
    extern __shared__ float smem[];
    float* Vs   = smem;                       // [512][VPAD]   34816
    float* Rs   = Vs   + 512 * VPAD;          // [L*KOUT]        512
    float* lps  = Rs   + 512;                 // [L*KOUT]        512
    float* mos  = lps  + 512;                 // [KOUT][MPAD]   1040
    float* vos  = mos  + KOUT * MPAD;         // [KOUT][MPAD]   1040
    float* fas  = vos  + KOUT * MPAD;         // [L]              32
    float* aouts  = fas + LN;                 // [KOUT]
    float* denoms = aouts + KOUT;
    float* lsgs   = denoms + KOUT;
    float* slogs  = lsgs + KOUT;

    const int n = blockIdx.x;
    const int t = threadIdx.x;                // t == l*16 + k  (512 pairs)
    const int l = t >> 4;
    const int k = t & 15;

    // ---- async DMA this n's V slice into LDS (bypasses VGPRs, ASYNCcnt) ----
    // Each lane owns one padded 64-float row: 16 x b128 transfers per lane.
    {
        const float* gsrc = V_ws + ((size_t)n * (LN * KOUT) + t) * ODIM;
        // low 32 bits of a flat pointer into LDS == LDS byte offset
        const unsigned lds_row = (unsigned)(uintptr_t)(Vs + t * VPAD);
        for (int q = 0; q < 16; ++q) {
            asm volatile("global_load_async_to_lds_b128 %0, %1, off"
                         :
                         : "v"(lds_row + 16u * q), "v"(gsrc + 4 * q)
                         : "memory");
        }
        asm volatile("s_wait_asynccnt 0x0" ::: "memory");
    }
    if (t < LN) fas[t] = fa_ws[(size_t)t * NTOK + n];
    Rs[t] = 1.0f / (float)KOUT;
    const int iters = *iters_p;
    __syncthreads();

    for (int it = 0; it < iters; ++it) {
        if (it > 0) {
            if (t < KOUT) {
                float s = 0.f;
                for (int o = 0; o < ODIM; ++o) s += logf(vos[t * MPAD + o] + EPSF);
                slogs[t] = s;
                const float a = aouts[t];
                lsgs[t] = fminf(a, 0.f) - log1pf(expf(-fabsf(a)));   // log_sigmoid
            }
            __syncthreads();
            // log_p for this (l,k)
            {
                const float* vrow = Vs + t * VPAD;
                const float* mo = mos + k * MPAD;
                const float* vo = vos + k * MPAD;
                float s = 0.f;
                for (int o = 0; o < ODIM; ++o) {
                    const float d = vrow[o] - mo[o];
                    s += d * d / (2.f * vo[o] + EPSF);
                }
                lps[t] = -s - 1.0f - HALF_PI - 0.5f * slogs[k];
            }
            __syncthreads();
            // R = softmax over k of (log_sigmoid(a_out) + log_p), per l
            if (t < LN) {
                float lg[KOUT];
                float mx = -3.0e38f;
                for (int kk = 0; kk < KOUT; ++kk) {
                    const float v = lsgs[kk] + lps[t * KOUT + kk];
                    lg[kk] = v; mx = fmaxf(mx, v);
                }
                float se = 0.f;
                for (int kk = 0; kk < KOUT; ++kk) { lg[kk] = expf(lg[kk] - mx); se += lg[kk]; }
                const float inv = 1.f / se;
                for (int kk = 0; kk < KOUT; ++kk) Rs[t * KOUT + kk] = lg[kk] * inv;
            }
            __syncthreads();
        }

        // denom[k], a_out[k] (sums over l)
        if (t < KOUT) {
            float dsum = 0.f, asum = 0.f;
            const float bk = beta_ign[t];
            for (int ll = 0; ll < LN; ++ll) {
                const float du = fas[ll] * Rs[ll * KOUT + t];
                dsum += du;
                asum += beta_use[ll] * du - bk * (fas[ll] - du);
            }
            denoms[t] = dsum + EPSF;
            aouts[t]  = asum;
        }
        __syncthreads();

        // mu_out[k][o] = sum_l D_use * V / denom
        for (int p = t; p < KOUT * ODIM; p += 512) {
            const int kk = p >> 6, o = p & 63;
            float s = 0.f;
            for (int ll = 0; ll < LN; ++ll)
                s += fas[ll] * Rs[ll * KOUT + kk] * Vs[(ll * KOUT + kk) * VPAD + o];
            mos[kk * MPAD + o] = s / denoms[kk];
        }
        __syncthreads();

        // var_out[k][o] = sum_l D_use * (V - mu)^2 / denom
        for (int p = t; p < KOUT * ODIM; p += 512) {
            const int kk = p >> 6, o = p & 63;
            const float mu = mos[kk * MPAD + o];
            float s = 0.f;
            for (int ll = 0; ll < LN; ++ll) {
                const float du = fas[ll] * Rs[ll * KOUT + kk];
                const float d  = Vs[(ll * KOUT + kk) * VPAD + o] - mu;
                s += du * d * d;
            }
            vos[kk * MPAD + o] = s / denoms[kk];
        }
        __syncthreads();
    }

    // ---- outputs: a_fin (N,KOUT) then mu_fin (N,KOUT,ODIM) ----
    if (t < KOUT) out[(size_t)n * KOUT + t] = aouts[t];
    for (int p = t; p < KOUT * ODIM; p += 512) {
        const int kk = p >> 6, o = p & 63;
        out[(size_t)NTOK * KOUT + ((size_t)n * KOUT + kk) * ODIM + o] = mos[kk * MPAD + o];
    }
}

// =====================================================================
extern "C" void kernel_launch(void* const* d_in, const int* in_sizes, int n_in,
                              void* d_out, int out_size, void* d_ws, size_t ws_size,
                              hipStream_t stream) {
    const float* x      = (const float*)d_in[0];
    const float* mask   = (const float*)d_in[1];
    const float* Wscore = (const float*)d_in[2];
    const float* Bscore = (const float*)d_in[3];
    const float* Wcap   = (const float*)d_in[4];
    const float* Bcap   = (const float*)d_in[5];
    const float* Wvote  = (const float*)d_in[6];
    const float* Bvote  = (const float*)d_in[7];
    const float* buse   = (const float*)d_in[8];
    const float* bign   = (const float*)d_in[9];
    const int*   iters  = (const int*)d_in[10];

    float* ws    = (float*)d_ws;
    float* fa_ws = ws;                 // L*N floats           (128 KB)
    float* V_ws  = ws + LN * NTOK;     // N*L*KOUT*ODIM floats (128 MB, L2-resident)

    k_stage1<<<dim3(LN * (NTOK / 16)), dim3(128), 0, stream>>>(
        x, mask, Wscore, Bscore, Wcap, Bcap, Wvote, Bvote, fa_ws, V_ws);

    const size_t lds_bytes =
        (512 * VPAD + 512 + 512 + 2 * KOUT * MPAD + LN + 4 * KOUT) * sizeof(float);
    k_route<<<dim3(NTOK), dim3(512), lds_bytes, stream>>>(
        fa_ws, V_ws, buse, bign, iters, (float*)d_out);
}